// CreativePositionalEncoding_8358006358352
// MI455X (gfx1250) — compile-verified
//
#include <hip/hip_runtime.h>

// ---------------------------------------------------------------------------
// CreativePositionalEncoding: out = x + concat(spatial_pe[h,w], pattern_pe[idx])
//
// Pure HBM-streaming op (0.125 FLOP/byte). Strategy:
//   - 1 block (256 thr = 8 wave32) per (b,h,w) pixel; 1 float4 per thread.
//   - Async global->LDS staging of the 2KB spatial row + 2KB gathered pattern
//     row (GLOBAL_LOAD_ASYNC_TO_LDS_B128, ASYNCcnt) overlapped with the
//     non-temporal b128 load of x.
//   - Non-temporal b128 load/store for the x/out streams (472 MB one-shot),
//     keeping the ~2MB PE tables resident in L2.
// ---------------------------------------------------------------------------

typedef float v4f __attribute__((ext_vector_type(4)));

#define D_MODEL       1024
#define HALF          (D_MODEL / 2)    // 512 floats per PE half
#define HALF_V4       (HALF / 4)       // 128 float4 chunks per half
#define MAX_GRID      30
#define NUM_PATTERNS  64
#define B_DIM         64
#define H_DIM         30
#define W_DIM         30
#define PIXELS        (B_DIM * H_DIM * W_DIM)   // 57600
#define THREADS       256                        // == D_MODEL/4 float4s

__global__ __launch_bounds__(THREADS)
void cpe_add_kernel(const float* __restrict__ x,
                    const int*   __restrict__ pidx,
                    const float* __restrict__ spe,   // [MAX_GRID, MAX_GRID, HALF]
                    const float* __restrict__ ppe,   // [NUM_PATTERNS, HALF]
                    float* __restrict__ out)
{
    __shared__ __align__(16) float tile[D_MODEL];  // [0..511] spatial, [512..1023] pattern

    const int pixel = blockIdx.x;                  // b*H*W + h*W + w
    const int hw    = pixel % (H_DIM * W_DIM);     // h*W + w (W == MAX_GRID)
    const int t     = threadIdx.x;                 // 0..255

    // Block-uniform pattern index -> scalar load path.
    const int idx = pidx[pixel] & (NUM_PATTERNS - 1);

    // Source of this thread's 16B chunk of the combined PE row.
    const float* pe_src;
    if (t < HALF_V4) {
        pe_src = spe + (size_t)hw  * HALF + (size_t)t * 4;           // spatial half
    } else {
        pe_src = ppe + (size_t)idx * HALF + (size_t)(t - HALF_V4) * 4; // pattern half
    }

    // Low 32 bits of the flat shared address == LDS byte offset for this wave.
    const unsigned lds_off = (unsigned)(size_t)(&tile[t * 4]);
    const unsigned long long gaddr = (unsigned long long)(const void*)pe_src;

    // Async copy 16B of PE row into LDS (tracked by ASYNCcnt).
    asm volatile("global_load_async_to_lds_b128 %0, %1, off"
                 :: "v"(lds_off), "v"(gaddr)
                 : "memory");

    // Stream x with a non-temporal 128-bit load while the async copy is in flight.
    const v4f* xin = (const v4f*)(x + (size_t)pixel * D_MODEL);
    v4f xv = __builtin_nontemporal_load(xin + t);

    // Drain this wave's async ops, then make LDS visible block-wide.
    asm volatile("s_wait_asynccnt 0" ::: "memory");
    __syncthreads();

    v4f pe = *(const v4f*)(&tile[t * 4]);
    v4f r  = xv + pe;

    v4f* o = (v4f*)(out + (size_t)pixel * D_MODEL);
    __builtin_nontemporal_store(r, o + t);
}

extern "C" void kernel_launch(void* const* d_in, const int* in_sizes, int n_in,
                              void* d_out, int out_size, void* d_ws, size_t ws_size,
                              hipStream_t stream)
{
    // setup_inputs order: x, pattern_indices, spatial_pe, pattern_pe
    const float* x    = (const float*)d_in[0];
    const int*   pidx = (const int*)  d_in[1];
    const float* spe  = (const float*)d_in[2];
    const float* ppe  = (const float*)d_in[3];
    float*       out  = (float*)d_out;

    (void)in_sizes; (void)n_in; (void)out_size; (void)d_ws; (void)ws_size;

    dim3 grid(PIXELS);
    dim3 block(THREADS);
    hipLaunchKernelGGL(cpe_add_kernel, grid, block, 0, stream, x, pidx, spe, ppe, out);
}